// DataTermLayer_11914239279555
// MI455X (gfx1250) — compile-verified
//
#include <hip/hip_runtime.h>

#define ALPHA 0.1f
#define BATCH 32
#define HH 512
#define WW 512

// Native clang vector types (HIP's float2/float4 are structs and are rejected
// by __builtin_nontemporal_load/store).
typedef __attribute__((ext_vector_type(2))) float v2f;
typedef __attribute__((ext_vector_type(4))) float v4f;

// Faithful reproduction of the reference's (buggy) bilinear sampler:
// x,y are the NORMALIZED coords in ~[-1,1]; corners are floor/clamped in that
// space, so all gathers hit I1[b, 0..2, 0..2] (L0-resident after first touch).
// Division by (W-1) is folded to a fused multiply by 2/(W-1): a single v_fma
// instead of a ~9-instruction IEEE divide expansion, keeping VALU time well
// under the HBM-stream time.
__device__ __forceinline__ float warp_sample(const float* __restrict__ I1b,
                                             float u, float v, int i, int j)
{
    constexpr float inv_w = 2.0f / (float)(WW - 1);
    constexpr float inv_h = 2.0f / (float)(HH - 1);
    const float newx = (float)j + 0.5f * u;
    const float newy = (float)i + 0.5f * v;
    const float x = __builtin_fmaf(newx, inv_w, -1.0f);
    const float y = __builtin_fmaf(newy, inv_h, -1.0f);

    float x0 = floorf(x), y0 = floorf(y);
    float x1 = x0 + 1.0f, y1 = y0 + 1.0f;
    x0 = fminf(fmaxf(x0, 0.0f), (float)(WW - 1));
    x1 = fminf(fmaxf(x1, 0.0f), (float)(WW - 1));
    y0 = fminf(fmaxf(y0, 0.0f), (float)(HH - 1));
    y1 = fminf(fmaxf(y1, 0.0f), (float)(HH - 1));

    const int x0i = (int)x0, x1i = (int)x1, y0i = (int)y0, y1i = (int)y1;
    const float Ia = I1b[y0i * WW + x0i];
    const float Ib = I1b[y1i * WW + x0i];
    const float Ic = I1b[y0i * WW + x1i];
    const float Id = I1b[y1i * WW + x1i];

    const float wa = (x1 - x) * (y1 - y);
    const float wb = (x1 - x) * (y - y0);
    const float wc = (x - x0) * (y1 - y);
    const float wd = (x - x0) * (y - y0);
    return wa * Ia + wb * Ib + wc * Ic + wd * Id;
}

__global__ __launch_bounds__(256)
void data_term_kernel(const float* __restrict__ I1,
                      const float* __restrict__ I2,
                      const float* __restrict__ flow,
                      float* __restrict__ out)
{
    const int tid = blockIdx.x * blockDim.x + threadIdx.x;
    const int npairs = BATCH * HH * (WW / 2);
    if (tid >= npairs) return;

    // two adjacent pixels (i, j0) and (i, j0+1)
    const int jj = tid & (WW / 2 - 1);      // 0..255
    const int i  = (tid >> 8) & (HH - 1);   // 0..511
    const int b  = tid >> 17;               // 0..31
    const int j0 = jj << 1;

    const size_t plane = (size_t)HH * WW;
    const float* __restrict__ I1b = I1 + (size_t)b * plane;
    const size_t pix = (size_t)i * WW + (size_t)j0;

    // Streaming single-use inputs: non-temporal wide loads.
    const v4f fl  = __builtin_nontemporal_load(
        reinterpret_cast<const v4f*>(flow + 2 * ((size_t)b * plane + pix)));
    const v2f i2v = __builtin_nontemporal_load(
        reinterpret_cast<const v2f*>(I2 + (size_t)b * plane + pix));

    // gfx1250 prefetch (global_prefetch_b8) a few cachelines ahead on the
    // heaviest stream.
    __builtin_prefetch(flow + 2 * ((size_t)b * plane + pix) + 1024, 0, 0);

    // I1 reused by gradients + gather -> regular (temporal) loads.
    const v2f c01 = *reinterpret_cast<const v2f*>(I1b + pix);   // (i,j0), (i,j0+1)

    const bool has_right = (j0 + 2) < WW;      // pixel1 has a dx neighbor
    const float c2 = has_right ? I1b[pix + 2] : 0.0f;

    const bool has_down = (i + 1) < HH;
    v2f d01 = c01;                              // values unused when !has_down
    if (has_down)
        d01 = *reinterpret_cast<const v2f*>(I1b + pix + WW);

    // tf.image_gradients: dy zero-padded at bottom, dx zero-padded at right.
    const float dy0 = has_down ? (d01.x - c01.x) : 0.0f;
    const float dy1 = has_down ? (d01.y - c01.y) : 0.0f;
    const float dx0 = c01.y - c01.x;            // j0 <= 510 always has right nbr
    const float dx1 = has_right ? (c2 - c01.y) : 0.0f;

    const float u0 = fl.x, v0 = fl.y;
    const float u1 = fl.z, v1 = fl.w;

    const float w0 = warp_sample(I1b, u0, v0, i, j0);
    const float w1 = warp_sample(I1b, u1, v1, i, j0 + 1);

    const float data0 = w0 - i2v.x;
    const float data1 = w1 - i2v.y;

    // Reference keeps the swapped names: u_next uses dy, v_next uses dx.
    v4f o;
    o.x = u0 - ALPHA * data0 * dy0;
    o.y = v0 - ALPHA * data0 * dx0;
    o.z = u1 - ALPHA * data1 * dy1;
    o.w = v1 - ALPHA * data1 * dx1;

    __builtin_nontemporal_store(
        o, reinterpret_cast<v4f*>(out + 2 * ((size_t)b * plane + pix)));
}

extern "C" void kernel_launch(void* const* d_in, const int* in_sizes, int n_in,
                              void* d_out, int out_size, void* d_ws, size_t ws_size,
                              hipStream_t stream)
{
    const float* I1   = (const float*)d_in[0];
    const float* I2   = (const float*)d_in[1];
    const float* flow = (const float*)d_in[2];
    float* out = (float*)d_out;

    const int npairs = BATCH * HH * (WW / 2);   // 4,194,304 threads, 2 px each
    const int block = 256;
    const int grid = (npairs + block - 1) / block;
    data_term_kernel<<<grid, block, 0, stream>>>(I1, I2, flow, out);
}